// GraphConvolution_5162550689989
// MI455X (gfx1250) — compile-verified
//
#include <hip/hip_runtime.h>

typedef __attribute__((ext_vector_type(2))) float v2f;
typedef __attribute__((ext_vector_type(4))) float v4f;
typedef __attribute__((ext_vector_type(8))) float v8f;
typedef __attribute__((ext_vector_type(4))) int   v4i;

#define BATCH   8192
#define DIN     1024
#define DOUT    256
#define MTOT    (BATCH * 4)     // 32768 flattened rows
#define KC      32              // K-chunk staged in LDS
#define LDSSTR  36              // padded row stride (floats): 16B-aligned rows, conflict-free b64 reads
#define CHUNK   (DOUT * LDSSTR) // floats per LDS buffer (36 KB)
#define WGROWS  128             // 8 waves x 16 rows

#if defined(__has_builtin)
#if __has_builtin(__builtin_amdgcn_global_load_async_to_lds_b128)
#define USE_ASYNC_LDS 1
#endif
#endif

typedef __attribute__((address_space(1))) v4i* gv4i_p;
typedef __attribute__((address_space(3))) v4i* lv4i_p;

__device__ __forceinline__ void wait_async0() {
#if defined(USE_ASYNC_LDS)
#if __has_builtin(__builtin_amdgcn_s_wait_asynccnt)
    __builtin_amdgcn_s_wait_asynccnt(0);
#else
    asm volatile("s_wait_asynccnt 0x0" ::: "memory");
#endif
#endif
}

// Stage W[:, kb:kb+KC] (256 rows x 32 floats) into one LDS buffer.
// 256 threads: thread t loads one float4 of row t>>3 per step, 8 steps.
__device__ __forceinline__ void stage_chunk(const float* __restrict__ W,
                                            float* __restrict__ buf,
                                            int tid, int kb) {
    const int row0 = tid >> 3;            // 0..31
    const int col4 = (tid & 7) * 4;       // 0,4,...,28
#pragma unroll
    for (int i = 0; i < 8; ++i) {
        const int o = row0 + i * 32;      // W row (= output column)
        const float* gp = W + (size_t)o * DIN + kb + col4;
        float* lp = &buf[o * LDSSTR + col4];
#if defined(USE_ASYNC_LDS)
        __builtin_amdgcn_global_load_async_to_lds_b128(
            (gv4i_p)gp, (lv4i_p)lp, /*offset=*/0, /*cpol=*/0);
#else
        *(v4f*)lp = *(const v4f*)gp;
#endif
    }
}

// out[r, o] = sum_m adj[n,m] * (feat_row[b*4+m] . W[o]) + bias[o],  r = b*4+n
// Plain GEMM G = F x W^T with WMMA f32 16x16x4; adj mix folded into the epilogue.
__global__ __launch_bounds__(256) void gcn_fused_wmma(
    const float* __restrict__ adj,    // [4,4]
    const float* __restrict__ feat,   // [MTOT, DIN]
    const float* __restrict__ W,      // [DOUT, DIN]
    const float* __restrict__ bias,   // [DOUT]
    float* __restrict__ out)          // [MTOT, DOUT]
{
    __shared__ float lds_w[2 * CHUNK];   // 72 KB, double-buffered

    const int tid  = threadIdx.x;
    const int wave = tid >> 5;
    const int lane = tid & 31;
    const int l16  = lane & 15;
    const int lh   = lane >> 4;

    const int mbase = blockIdx.x * WGROWS + wave * 16;

    // adj (uniform -> SGPRs)
    float adjS[16];
#pragma unroll
    for (int i = 0; i < 16; ++i) adjS[i] = adj[i];

    // bias fragment: column o = nt*16 + l16
    float biasF[16];
#pragma unroll
    for (int nt = 0; nt < 16; ++nt) biasF[nt] = bias[nt * 16 + l16];

    v8f acc[16];
#pragma unroll
    for (int nt = 0; nt < 16; ++nt) {
        v8f z = {0.f, 0.f, 0.f, 0.f, 0.f, 0.f, 0.f, 0.f};
        acc[nt] = z;
    }

    // A-fragment base: row = mbase + l16, K offset lh*2 (lanes 0-15: K0/K1, 16-31: K2/K3)
    const float* aPtr = feat + (size_t)(mbase + l16) * DIN + lh * 2;

    // Prologue: stage first chunk into buffer 0.
    stage_chunk(W, &lds_w[0], tid, 0);

    int ib = 0;
    for (int kb = 0; kb < DIN; kb += KC, ib ^= 1) {
        wait_async0();          // my staging ops done (async path only)
        __syncthreads();        // everyone's staging visible; prev compute done

        // Kick off next chunk into the other buffer (overlaps with compute below).
        if (kb + KC < DIN)
            stage_chunk(W, &lds_w[(ib ^ 1) * CHUNK], tid, kb + KC);

        const float* cur = &lds_w[ib * CHUNK];

        // ---- K-chunk compute: 8 steps of K=4, 16 N-tiles each ----
        for (int kc = 0; kc < KC; kc += 4) {
            v2f a = *(const v2f*)(aPtr + kb + kc);

            v2f bf[16];
#pragma unroll
            for (int nt = 0; nt < 16; ++nt)
                bf[nt] = *(const v2f*)&cur[(nt * 16 + l16) * LDSSTR + kc + lh * 2];

#pragma unroll
            for (int nt = 0; nt < 16; ++nt)
                acc[nt] = __builtin_amdgcn_wmma_f32_16x16x4_f32(
                    /*neg_a=*/false, a, /*neg_b=*/false, bf[nt],
                    /*c_mod=*/(short)0, acc[nt],
                    /*reuse_a=*/false, /*reuse_b=*/false);
        }
    }

    // ---- epilogue: adj mix within batch groups of 4 rows, + bias, store ----
    // lanes 0-15 hold rows mbase+0..7 in acc v0..7; lanes 16-31 hold rows mbase+8..15.
    const int rbase = mbase + lh * 8;
#pragma unroll
    for (int nt = 0; nt < 16; ++nt) {
        v8f c = acc[nt];
        float o[8];
#pragma unroll
        for (int g = 0; g < 2; ++g)
#pragma unroll
            for (int n = 0; n < 4; ++n)
                o[g * 4 + n] = adjS[n * 4 + 0] * c[g * 4 + 0]
                             + adjS[n * 4 + 1] * c[g * 4 + 1]
                             + adjS[n * 4 + 2] * c[g * 4 + 2]
                             + adjS[n * 4 + 3] * c[g * 4 + 3]
                             + biasF[nt];
        float* op = out + (size_t)rbase * DOUT + nt * 16 + l16;
#pragma unroll
        for (int v = 0; v < 8; ++v)
            op[v * DOUT] = o[v];
    }
}

extern "C" void kernel_launch(void* const* d_in, const int* in_sizes, int n_in,
                              void* d_out, int out_size, void* d_ws, size_t ws_size,
                              hipStream_t stream) {
    const float* adj  = (const float*)d_in[0];   // [4,4]
    const float* feat = (const float*)d_in[1];   // [8192,4,1024]
    const float* W    = (const float*)d_in[2];   // [256,1024]
    const float* bias = (const float*)d_in[3];   // [256]
    float* out = (float*)d_out;                  // [8192,4,256]

    dim3 grid(MTOT / WGROWS);   // 256 workgroups
    dim3 block(256);            // 8 waves (wave32)
    hipLaunchKernelGGL(gcn_fused_wmma, grid, block, 0, stream,
                       adj, feat, W, bias, out);
}